// PackedMultiSubtableNgramTableBank_1717986918572
// MI455X (gfx1250) — compile-verified
//
#include <hip/hip_runtime.h>
#include <hip/hip_bf16.h>

// ---- problem constants (match reference) ----
#define B_      4
#define SUB_    2
#define T_      2048        // 2^11
#define R_      128
#define ALPHA_  16
#define MEM_    16
#define RV2     256         // 16^2
#define RV3     4096        // 16^3
#define TV2     (R_ * RV2)  // 32768
#define TV3     (R_ * RV3)  // 524288
#define SUFFIX_ (R_ * MEM_) // 2048
#define FLAT_   (2 * SUFFIX_) // 4096

typedef float v4f __attribute__((ext_vector_type(4)));

// CDNA5 async global->LDS copy (ASYNCcnt path). Guarded so a differing builtin
// name just degrades to a synchronous copy instead of breaking the build.
#if defined(__HIP_DEVICE_COMPILE__) && __has_builtin(__builtin_amdgcn_global_load_async_to_lds_b32)
#define ASYNC_LDS 1
#else
#define ASYNC_LDS 0
#endif

typedef __attribute__((address_space(1))) int gint_t;
typedef __attribute__((address_space(3))) int lint_t;

__device__ __forceinline__ void stage_b32(int* lds_dst, const int* gsrc) {
#if ASYNC_LDS
  __builtin_amdgcn_global_load_async_to_lds_b32(
      (gint_t*)gsrc,      // global source (addrspacecast generic->AS1, const stripped)
      (lint_t*)lds_dst,   // LDS destination (addrspacecast generic->AS3)
      /*offset=*/0, /*cpol=*/0);
#else
  *lds_dst = *gsrc;
#endif
}

__device__ __forceinline__ void wait_async_lds() {
#if ASYNC_LDS
#if __has_builtin(__builtin_amdgcn_s_wait_asynccnt)
  __builtin_amdgcn_s_wait_asynccnt(0);
#else
  asm volatile("s_wait_asynccnt 0" ::: "memory");
#endif
#endif
}

// One block per output row (b,s,p). 256 threads (8 wave32), each thread emits
// four b128 stores -> 1024 float4 = the full FLAT_=4096-float row, contiguous.
__global__ __launch_bounds__(256)
void ngram_table_gather(const int* __restrict__ codes,
                        const float* __restrict__ w2,
                        const float* __restrict__ w3,
                        float* __restrict__ out)
{
  __shared__ int cds[3][R_];  // rows p, p-1, p-2 of route codes for this (b,s)

  const int bid = blockIdx.x;
  const int p   = bid & (T_ - 1);
  const int sb  = bid >> 11;           // b*SUB + s
  const int s   = sb & (SUB_ - 1);
  const int tid = threadIdx.x;

  const int* crow = codes + (size_t)sb * T_ * R_;

  // ---- async-stage the (up to) 3 code rows into LDS ----
  {
    const int r = tid & (R_ - 1);
    const int k = tid >> 7;            // threads 0..127 -> row p, 128..255 -> row p-1
    if (p - k >= 0) stage_b32(&cds[k][r], crow + (size_t)(p - k) * R_ + r);
    else            cds[k][r] = 0;
    if (tid < R_) {
      if (p >= 2)   stage_b32(&cds[2][r], crow + (size_t)(p - 2) * R_ + r);
      else          cds[2][r] = 0;
    }
  }
  wait_async_lds();     // each wave drains its own ASYNCcnt
  __syncthreads();      // then cross-wave visibility of LDS

  v4f* outrow = (v4f*)(out + ((size_t)sb * T_ + p) * FLAT_);

#pragma unroll
  for (int j = 0; j < 4; ++j) {
    const int e = j * 256 + tid;       // float4 slot in the 4096-float row
    const int r = (e >> 2) & (R_ - 1); // route
    const int q = e & 3;               // which 16B quarter of the 64B table row
    v4f v = (v4f)0.0f;
    if (e < 512) {                     // bigram section (first SUFFIX_)
      if (p >= 1) {
        const int addr = cds[1][r] + ALPHA_ * cds[0][r];
        const float* wrow = w2 + ((size_t)s * TV2 + (size_t)r * RV2 + (size_t)addr) * MEM_;
        v = *(const v4f*)(wrow + q * 4);   // L2-resident gather (w2 = 4 MB)
      }
    } else {                           // trigram section (second SUFFIX_)
      if (p >= 2) {
        const int addr = cds[2][r] + ALPHA_ * cds[1][r] + (ALPHA_ * ALPHA_) * cds[0][r];
        const float* wrow = w3 + ((size_t)s * TV3 + (size_t)r * RV3 + (size_t)addr) * MEM_;
        v = *(const v4f*)(wrow + q * 4);   // L2-resident gather (w3 = 64 MB < 192 MB L2)
      }
    }
    // Non-temporal b128 store: 256 MB output stream must not evict the
    // 68 MB of tables living in L2.
    __builtin_nontemporal_store(v, outrow + e);
  }
}

extern "C" void kernel_launch(void* const* d_in, const int* in_sizes, int n_in,
                              void* d_out, int out_size, void* d_ws, size_t ws_size,
                              hipStream_t stream) {
  (void)in_sizes; (void)n_in; (void)out_size; (void)d_ws; (void)ws_size;
  const int*   codes = (const int*)d_in[0];   // [B,SUB,T,R] int32
  const float* w2    = (const float*)d_in[1]; // [SUB,TV2,MEM] f32
  const float* w3    = (const float*)d_in[2]; // [SUB,TV3,MEM] f32
  float*       out   = (float*)d_out;         // [B,SUB,T,FLAT] f32

  ngram_table_gather<<<dim3(B_ * SUB_ * T_), dim3(256), 0, stream>>>(codes, w2, w3, out);
}